// RagContrastive_56882546868663
// MI455X (gfx1250) — compile-verified
//
#include <hip/hip_runtime.h>

// RAG contrastive loss for MI455X (gfx1250, wave32).
//   embeddings: (1,16,1024,1024) f32 channel-major   weights: (16384,) f32
//   sp_seg:     (1,1,1024,1024) i32 in [0,4096)      edges:   (2,16384) i32
// Memory-bound: ~136 MB mandatory HBM traffic -> ~5.9us floor @ 23.3 TB/s.

#define C_SEG      4096
#define E_DIM      16
#define DELTA_VAR  0.5f
#define DELTA_DIST 1.5f

typedef __attribute__((ext_vector_type(2))) float f32x2;
typedef __attribute__((ext_vector_type(8))) float f32x8;

// ---------------------------------------------------------------- init
__global__ void init_ws_kernel(float* __restrict__ ws, int n) {
    int i = blockIdx.x * blockDim.x + threadIdx.x;
    if (i < n) ws[i] = 0.0f;
}

// ------------------------------------------- segment sums (LDS-privatized)
// Per-block private accumulators in LDS (C*17 floats = 272KB; CDNA5 WGP has
// 320KB LDS), ds_add_f32 per pixel, one global_atomic_add_f32 flush per block.
__global__ __launch_bounds__(256) void accum_kernel(
    const float* __restrict__ emb, const int* __restrict__ seg,
    float* __restrict__ gsum, float* __restrict__ gcnt, int P) {
    extern __shared__ float lds[];
    float* lsum = lds;                   // C_SEG * E_DIM
    float* lcnt = lds + C_SEG * E_DIM;   // C_SEG
    for (int i = threadIdx.x; i < C_SEG * E_DIM + C_SEG; i += blockDim.x) lds[i] = 0.0f;
    __syncthreads();

    const int nquad  = P >> 2;
    const int stride = gridDim.x * blockDim.x;
    for (int q = blockIdx.x * blockDim.x + threadIdx.x; q < nquad; q += stride) {
        const int  p  = q << 2;
        const int4 s4 = *reinterpret_cast<const int4*>(seg + p);
        atomicAdd(&lcnt[s4.x], 1.0f);
        atomicAdd(&lcnt[s4.y], 1.0f);
        atomicAdd(&lcnt[s4.z], 1.0f);
        atomicAdd(&lcnt[s4.w], 1.0f);
#pragma unroll
        for (int e = 0; e < E_DIM; ++e) {   // channel-major: coalesced float4 per wave
            const float4 v = *reinterpret_cast<const float4*>(emb + (size_t)e * P + p);
            atomicAdd(&lsum[s4.x * E_DIM + e], v.x);
            atomicAdd(&lsum[s4.y * E_DIM + e], v.y);
            atomicAdd(&lsum[s4.z * E_DIM + e], v.z);
            atomicAdd(&lsum[s4.w * E_DIM + e], v.w);
        }
    }
    __syncthreads();
    for (int i = threadIdx.x; i < C_SEG * E_DIM; i += blockDim.x) {
        const float v = lsum[i];
        if (v != 0.0f) atomicAdd(&gsum[i], v);
    }
    for (int i = threadIdx.x; i < C_SEG; i += blockDim.x) {
        const float v = lcnt[i];
        if (v != 0.0f) atomicAdd(&gcnt[i], v);
    }
}

// ----------------------------------------------------- means = sums / n
__global__ void finalize_means_kernel(float* __restrict__ sums, float* __restrict__ counts) {
    const int c = blockIdx.x * blockDim.x + threadIdx.x;
    if (c >= C_SEG) return;
    const float n = fmaxf(counts[c], 1.0f);
    counts[c] = n;                         // keep clamped n for the intra term
    const float inv = 1.0f / n;
#pragma unroll
    for (int e = 0; e < E_DIM; ++e) sums[c * E_DIM + e] *= inv;
}

// ---------------------------------------- intra term (means staged in LDS)
__global__ __launch_bounds__(256) void intra_kernel(
    const float* __restrict__ emb, const int* __restrict__ seg,
    const float* __restrict__ means, const float* __restrict__ nvals,
    float* __restrict__ acc, int P) {
    extern __shared__ float lds[];
    float* lmean = lds;                    // 256KB: full means table resident
    float* linv  = lds + C_SEG * E_DIM;    // 16KB : 1/n per segment
    for (int i = threadIdx.x; i < C_SEG * E_DIM; i += blockDim.x) lmean[i] = means[i];
    for (int i = threadIdx.x; i < C_SEG; i += blockDim.x) linv[i] = 1.0f / nvals[i];
    __syncthreads();

    float local = 0.0f;
    const int nquad  = P >> 2;
    const int stride = gridDim.x * blockDim.x;
    for (int q = blockIdx.x * blockDim.x + threadIdx.x; q < nquad; q += stride) {
        const int  p  = q << 2;
        const int4 s4 = *reinterpret_cast<const int4*>(seg + p);
        float d0 = 0.f, d1 = 0.f, d2 = 0.f, d3 = 0.f;
#pragma unroll
        for (int e = 0; e < E_DIM; ++e) {
            const float4 v = *reinterpret_cast<const float4*>(emb + (size_t)e * P + p);
            d0 += lmean[s4.x * E_DIM + e] * v.x;
            d1 += lmean[s4.y * E_DIM + e] * v.y;
            d2 += lmean[s4.z * E_DIM + e] * v.z;
            d3 += lmean[s4.w * E_DIM + e] * v.w;
        }
        local += fmaxf(1.0f - d0 - DELTA_VAR, 0.0f) * linv[s4.x];
        local += fmaxf(1.0f - d1 - DELTA_VAR, 0.0f) * linv[s4.y];
        local += fmaxf(1.0f - d2 - DELTA_VAR, 0.0f) * linv[s4.z];
        local += fmaxf(1.0f - d3 - DELTA_VAR, 0.0f) * linv[s4.w];
    }
#pragma unroll
    for (int off = 16; off > 0; off >>= 1) local += __shfl_down(local, off, 32);
    if ((threadIdx.x & 31) == 0) atomicAdd(&acc[0], local * (1.0f / (float)C_SEG));
}

// ------------------- inter term: 16-edge tiles via v_wmma_f32_16x16x4_f32
// D = A(16x16) * B(16x16) chained over four K=4 slices; we need only diag(D):
//   diag i<8  -> lane i,     accumulator VGPR i
//   diag i>=8 -> lane i+16,  accumulator VGPR i-8
__global__ __launch_bounds__(256) void inter_wmma_kernel(
    const float* __restrict__ means, const int* __restrict__ edges,
    const float* __restrict__ w, float* __restrict__ acc, int ne) {
    const int lane  = threadIdx.x & 31;
    const int wave  = (blockIdx.x * blockDim.x + threadIdx.x) >> 5;
    const int nwave = (gridDim.x * blockDim.x) >> 5;
    const int ntile = ne >> 4;
    const int r     = lane & 15;               // A row (M) / B col (N) owned by lane
    const int kbase = (lane >= 16) ? 2 : 0;    // K pair per 32-bit operand layout

    float local = 0.0f;
    for (int t = wave; t < ntile; t += nwave) {     // uniform per wave -> EXEC all-1s
        const float* Arow = means + (size_t)edges[t * 16 + r]      * E_DIM; // means[u]
        const float* Brow = means + (size_t)edges[ne + t * 16 + r] * E_DIM; // means[v]
        f32x8 c = {};
#pragma unroll
        for (int kk = 0; kk < 4; ++kk) {
            const int k0 = kk * 4 + kbase;
            f32x2 a; a.x = Arow[k0]; a.y = Arow[k0 + 1];
            f32x2 b; b.x = Brow[k0]; b.y = Brow[k0 + 1];
            c = __builtin_amdgcn_wmma_f32_16x16x4_f32(
                    /*neg_a=*/false, a, /*neg_b=*/false, b,
                    /*c_mod=*/(short)0, c, /*reuse_a=*/false, /*reuse_b=*/false);
        }
        float diag; int eidx;
        if (lane < 8)        { diag = c[lane];      eidx = lane; }
        else if (lane >= 24) { diag = c[lane - 24]; eidx = lane - 16; }
        else                 { diag = 0.0f;         eidx = -1; }
        if (eidx >= 0) {
            const float d = 1.0f - diag;
            local += fmaxf(DELTA_DIST - d * w[t * 16 + eidx], 0.0f);
        }
    }
#pragma unroll
    for (int off = 16; off > 0; off >>= 1) local += __shfl_down(local, off, 32);
    if (lane == 0) atomicAdd(&acc[1], local * (1.0f / (float)ne));
}

// ---------------------------------------------------------------- combine
__global__ void final_kernel(const float* __restrict__ acc, float* __restrict__ out) {
    out[0] = acc[0] + acc[1];   // BETA*intra + ALPHA*inter (both 1.0)
}

extern "C" void kernel_launch(void* const* d_in, const int* in_sizes, int n_in,
                              void* d_out, int out_size, void* d_ws, size_t ws_size,
                              hipStream_t stream) {
    const float* emb     = (const float*)d_in[0];
    const float* weights = (const float*)d_in[1];
    const int*   seg     = (const int*)d_in[2];
    const int*   edges   = (const int*)d_in[3];
    const int P  = in_sizes[2];        // pixels (H*W)
    const int NE = in_sizes[1];        // edges

    // ws layout (floats): [0,C*E) sums->means | [C*E,C*E+C) counts->n | 2 accumulators
    float* ws     = (float*)d_ws;
    float* sums   = ws;
    float* counts = ws + C_SEG * E_DIM;
    float* acc    = counts + C_SEG;
    const int n_init = C_SEG * E_DIM + C_SEG + 2;

    const size_t lds_bytes = (size_t)(C_SEG * E_DIM + C_SEG) * sizeof(float); // 272KB < 320KB/WGP
    (void)hipFuncSetAttribute((const void*)accum_kernel,
                              hipFuncAttributeMaxDynamicSharedMemorySize, (int)lds_bytes);
    (void)hipFuncSetAttribute((const void*)intra_kernel,
                              hipFuncAttributeMaxDynamicSharedMemorySize, (int)lds_bytes);

    init_ws_kernel<<<(n_init + 255) / 256, 256, 0, stream>>>(ws, n_init);
    accum_kernel<<<120, 256, lds_bytes, stream>>>(emb, seg, sums, counts, P);
    finalize_means_kernel<<<C_SEG / 256, 256, 0, stream>>>(sums, counts);
    intra_kernel<<<120, 256, lds_bytes, stream>>>(emb, seg, sums, counts, acc, P);
    inter_wmma_kernel<<<32, 256, 0, stream>>>(sums, edges, weights, acc, NE);
    final_kernel<<<1, 1, 0, stream>>>(acc, (float*)d_out);
}